// DWTExtractor_64192581206249
// MI455X (gfx1250) — compile-verified
//
#include <hip/hip_runtime.h>
#include <hip/hip_bf16.h>

// CDNA5 / gfx1250. Wave32. WMMA f32 16x16x4 executes the banded separable
// operators (Haar DWT, Coif1 DWT ∘ antialiased-linear-resize) as tile GEMMs:
//   O(16x16) = V'(16x64) * X(64x64) * H'^T(64x16)
// Roofline: 335 MB total traffic @ 23.3 TB/s ≈ 14 us; WMMA compute overlaps.
// All WMMA-reaching control flow is forced wave-uniform via readfirstlane so
// EXEC is all-ones at every v_wmma (ISA §7.12 requirement) and loops are
// scalar (s_cmp/s_cbranch), not exec-masked.

typedef __attribute__((ext_vector_type(2))) float v2f;
typedef __attribute__((ext_vector_type(8))) float v8f;

#define WIN   64   // input window per tile (rows and cols)
#define XSTR  72   // LDS row stride for X window (bank-conflict padding)
#define USTR  73   // LDS row stride for U (stage-1 result)
#define WSTR  17   // LDS row stride for banded weight rows (16 used)
#define NTHREADS 192

// Build one banded operator row for output index o of transform t:
//   t=0 haar-lo, t=1 haar-hi, t=2 coif-lo∘resize, t=3 coif-hi∘resize
// Writes 16 weights (LDS) and the absolute input base (clamped to [0,1008]).
__device__ inline void build_row(int t, int o, float* __restrict__ wrow,
                                 int* __restrict__ base_out) {
  #pragma unroll
  for (int k = 0; k < 16; ++k) wrow[k] = 0.0f;

  if (t < 2) {
    // Haar, L=2: y[k] = fr0*x[2k] + fr1*x[2k+1]; padding never reached.
    const float r2 = 0.70710678118654752f;
    int braw = 2 * o;
    int base = braw > 1008 ? 1008 : braw;
    int off  = braw - base;
    wrow[off]     = r2;
    wrow[off + 1] = (t == 0) ? r2 : -r2;
    *base_out = base;
  } else {
    // Coif1 (L=6) DWT with half-sample symmetric padding, composed with
    // jax.image.resize linear (antialias, scale = 512/514, normalized).
    const float c0 = -0.01565572813546454f;
    const float c1 = -0.0727326195128539f;
    const float c2 =  0.38486484686420286f;
    const float c3 =  0.8525720202122554f;
    const float c4 =  0.3378976624578092f;
    const float c5 = -0.0727326195128539f;
    // reversed filters: fr_lo[j] = lo[5-j]; fr_hi[j] = (-1)^j * c[j]
    float fr[6];
    if (t == 2) { fr[0]=c5; fr[1]=c4; fr[2]=c3; fr[3]=c2; fr[4]=c1; fr[5]=c0; }
    else        { fr[0]=c0; fr[1]=-c1; fr[2]=c2; fr[3]=-c3; fr[4]=c4; fr[5]=-c5; }

    const float sc  = 512.0f / 514.0f;   // kernel scale (antialias, <1)
    const float rad = 514.0f / 512.0f;   // support radius = 1/sc
    float sf = (o + 0.5f) * rad - 0.5f;  // sample coordinate in 514-space
    int ilo = (int)ceilf(sf - rad);
    int ihi = (int)floorf(sf + rad);
    if (ilo < 0)   ilo = 0;
    if (ihi > 513) ihi = 513;
    int ntap = ihi - ilo + 1;            // 2 or 3

    float rw[3]; float tot = 0.0f;
    #pragma unroll
    for (int k = 0; k < 3; ++k) {
      float w = 0.0f;
      if (k < ntap) w = fmaxf(1.0f - fabsf((float)(ilo + k) - sf) * sc, 0.0f);
      rw[k] = w; tot += w;
    }
    float rtot = 1.0f / tot;

    int braw = 2 * ilo - 4;
    int base = braw < 0 ? 0 : (braw > 1008 ? 1008 : braw);
    #pragma unroll
    for (int k = 0; k < 3; ++k) {
      if (k < ntap) {
        float rwk = rw[k] * rtot;
        #pragma unroll
        for (int j = 0; j < 6; ++j) {
          int m = 2 * (ilo + k) + j - 4;       // DWT input index
          if (m < 0)     m = -1 - m;           // symmetric reflection
          if (m >= 1024) m = 2047 - m;
          wrow[m - base] += rwk * fr[j];       // folded band stays within 16
        }
      }
    }
    *base_out = base;
  }
}

// Branchless banded-row fetch: wrap index into the 16-wide row, select 0
// outside the band. No EXEC manipulation (keeps WMMA preconditions trivial).
__device__ inline float band_get(const float* __restrict__ wrow, int r) {
  unsigned ur = (unsigned)r;
  float v = wrow[ur & 15u];
  return (ur < 16u) ? v : 0.0f;
}

__global__ __launch_bounds__(NTHREADS)
void dwt6_kernel(const float* __restrict__ x, float* __restrict__ out) {
  __shared__ float Xs[WIN * XSTR];        // 64x64 input window
  __shared__ float Us[4 * 16 * USTR];     // stage-1 results, 4 transforms x 16x64
  __shared__ float Ws[8 * 16 * WSTR];     // banded weight rows (4 transforms x 2 axes)
  __shared__ int   Bs[8 * 16];            // absolute base per row
  __shared__ int   Kl[8], Kh[8];          // window-relative K ranges per set
  __shared__ int   W0[2];                 // window base: [0]=vertical [1]=horizontal

  const int tid  = threadIdx.x;
  const int lane = tid & 31;
  const int l16  = lane & 15;
  const int hi   = lane >> 4;
  const int wvu  = __builtin_amdgcn_readfirstlane(tid >> 5);  // scalar wave id

  const int bid = blockIdx.x;
  const int img = bid >> 10;
  const int r0  = ((bid >> 5) & 31) * 16;
  const int c0  = (bid & 31) * 16;

  // ---- Phase 1: banded operator rows. Set s = axis*4 + transform. ----
  if (tid < 128) {
    int s = tid >> 4;                     // 0..3 vertical, 4..7 horizontal
    int i = tid & 15;
    int o = ((s >> 2) ? c0 : r0) + i;
    build_row(s & 3, o, &Ws[(s * 16 + i) * WSTR], &Bs[s * 16 + i]);
  }
  __syncthreads();

  // ---- Phase 2: base min/max per set, then window bases. ----
  if (tid < 8) {
    int s = tid;
    int mn = Bs[s * 16], mx = mn;
    for (int i = 1; i < 16; ++i) {
      int b = Bs[s * 16 + i];
      mn = min(mn, b); mx = max(mx, b);
    }
    Kl[s] = mn; Kh[s] = mx;               // absolute, fixed up below
  }
  __syncthreads();
  if (tid < 2) {
    int mn = Kl[tid * 4];
    for (int s = 1; s < 4; ++s) mn = min(mn, Kl[tid * 4 + s]);
    W0[tid] = min(mn, 1024 - WIN);        // union span <= 48, so WIN=64 covers
  }
  __syncthreads();
  // ---- Phase 3: K ranges (threads 0..7) + X window load (all threads). ----
  if (tid < 8) {
    int w0 = W0[tid >> 2];
    int lo = (Kl[tid] - w0) & ~3;
    int hh = ((Kh[tid] + 16 - w0) + 3) & ~3;   // <= 64 by construction
    Kl[tid] = lo; Kh[tid] = hh;
  }
  {
    const float* src = x + (size_t)img * (1024u * 1024u);
    int w0v = W0[0], w0h = W0[1];
    for (int idx = tid; idx < WIN * WIN; idx += NTHREADS) {
      int r = idx >> 6, c = idx & 63;
      Xs[r * XSTR + c] = src[(size_t)(w0v + r) * 1024 + (w0h + c)];
    }
  }
  __syncthreads();

  const int w0v = __builtin_amdgcn_readfirstlane(W0[0]);
  const int w0h = __builtin_amdgcn_readfirstlane(W0[1]);

  // ---- Stage 1: U_t = V'_t * X, t = {haarLo, haarHi, coifLo, coifHi}. ----
  // 16 GEMM units = 4 transforms x 4 N-tiles, spread over the 6 waves.
  for (int u = wvu; u < 16; u += 6) {
    const int t   = u >> 2;
    const int n0  = (u & 3) * 16;
    const int klo = __builtin_amdgcn_readfirstlane(Kl[t]);   // scalar bounds
    const int khi = __builtin_amdgcn_readfirstlane(Kh[t]);
    const float* wrow = &Ws[(t * 16 + l16) * WSTR];
    const int boff = Bs[t * 16 + l16] - w0v;                 // per-lane band start
    const float* xcol = &Xs[n0 + l16];

    v8f acc = {0.f,0.f,0.f,0.f,0.f,0.f,0.f,0.f};
    for (int k0 = klo; k0 < khi; k0 += 4) {
      int p = k0 + 2 * hi;
      // A fragment (16x4): lanes 0-15 rows, VGPR0<->K{0,2}, VGPR1<->K{1,3}
      v2f a; a.x = band_get(wrow, p - boff); a.y = band_get(wrow, p - boff + 1);
      // B fragment (4x16): lanes index N; VGPR0<->K{0,2}, VGPR1<->K{1,3}
      v2f b; b.x = xcol[p * XSTR]; b.y = xcol[(p + 1) * XSTR];
      acc = __builtin_amdgcn_wmma_f32_16x16x4_f32(
                false, a, false, b, (short)0, acc, false, false);
    }
    float* urow = &Us[(t * 16 + hi * 8) * USTR + (n0 + l16)];
    #pragma unroll
    for (int v = 0; v < 8; ++v)                 // D: VGPR v <-> row v(+8)
      urow[v * USTR] = acc[v];
  }
  __syncthreads();

  // ---- Stage 2: O = U * H'^T -> 6 output channels, one GEMM per wave. ----
  {
    const int wavelet = wvu / 3;           // 0 haar, 1 coif
    const int ch      = wvu % 3;           // 0 cH, 1 cV, 2 cD
    const int uf = (ch == 1) ? 0 : 1;      // vertical filter: hi except cV
    const int hf = (ch == 0) ? 0 : 1;      // horizontal filter: lo for cH
    const int tu = wavelet * 2 + uf;
    const int hs = 4 + wavelet * 2 + hf;   // horizontal operator set
    const int klo = __builtin_amdgcn_readfirstlane(Kl[hs]);
    const int khi = __builtin_amdgcn_readfirstlane(Kh[hs]);
    const float* wrow = &Ws[(hs * 16 + l16) * WSTR];
    const int boff = Bs[hs * 16 + l16] - w0h;
    const float* urow = &Us[(tu * 16 + l16) * USTR];

    v8f acc = {0.f,0.f,0.f,0.f,0.f,0.f,0.f,0.f};
    for (int k0 = klo; k0 < khi; k0 += 4) {
      int p = k0 + 2 * hi;
      v2f a; a.x = urow[p]; a.y = urow[p + 1];          // A = U (16x64)
      // B[q][j] = H'[j][q] -> banded horizontal operator row
      v2f b; b.x = band_get(wrow, p - boff); b.y = band_get(wrow, p - boff + 1);
      acc = __builtin_amdgcn_wmma_f32_16x16x4_f32(
                false, a, false, b, (short)0, acc, false, false);
    }
    const int chan = wavelet * 3 + ch;
    float* dst = out + (((size_t)img * 6 + chan) * 512 + (r0 + hi * 8)) * 512
                     + c0 + l16;
    #pragma unroll
    for (int v = 0; v < 8; ++v)
      dst[(size_t)v * 512] = acc[v];
  }
}

extern "C" void kernel_launch(void* const* d_in, const int* in_sizes, int n_in,
                              void* d_out, int out_size, void* d_ws, size_t ws_size,
                              hipStream_t stream) {
  const float* x = (const float*)d_in[0];
  float* out = (float*)d_out;
  int B = in_sizes[0] >> 20;               // elements / (1024*1024) images
  dim3 grid((unsigned)(B * 32 * 32));      // one block per (image, 16x16 tile)
  dwt6_kernel<<<grid, NTHREADS, 0, stream>>>(x, out);
}